// ASTGATEncoder_27642409517290
// MI455X (gfx1250) — compile-verified
//
#include <hip/hip_runtime.h>
#include <math.h>

#define NHEADS 4
#define F_IN   256
#define HID    256
#define OUTD   512
#define SLOPE  0.2f
#define LN_EPS 1e-5f

#define KCHUNK 64          // K-chunk staged per double-buffer step
#define BSTRIDE 68         // padded LDS row stride (floats): bank = (4*col + k) % 64, conflict-free

typedef float v2f __attribute__((ext_vector_type(2)));
typedef float v8f __attribute__((ext_vector_type(8)));

// ---------- helpers ----------

// Monotone order-preserving float->uint mapping so unsigned atomicMax == float max.
// memset(0) == smallest key (below -inf), so no separate -inf init pass is needed.
__device__ __forceinline__ unsigned flipf(float f) {
  unsigned u = __float_as_uint(f);
  return (u & 0x80000000u) ? ~u : (u | 0x80000000u);
}
__device__ __forceinline__ float unflipf(unsigned u) {
  u = (u & 0x80000000u) ? (u & 0x7FFFFFFFu) : ~u;
  return __uint_as_float(u);
}

// CDNA5 async global->LDS copy (ASYNCcnt-tracked); 16B per lane.
// LDS dest address passed as 32-bit byte offset in a VGPR (low 32 bits of the
// flat LDS-aperture address are the LDS offset on gfx1250).
__device__ __forceinline__ void async_copy16(const float* g, unsigned lds_addr) {
  asm volatile("global_load_async_to_lds_b128 %0, %1, off"
               :: "v"(lds_addr), "v"(g) : "memory");
}
__device__ __forceinline__ void wait_async_le(void) {   // allow the next chunk's 4 in flight
  asm volatile("s_wait_asynccnt 0x4" ::: "memory");
}
__device__ __forceinline__ void wait_async_0(void) {
  asm volatile("s_wait_asynccnt 0x0" ::: "memory");
}

__global__ void fill_zero_u32(unsigned* __restrict__ p, long n) {
  long i = (long)blockIdx.x * blockDim.x + threadIdx.x;
  if (i < n) p[i] = 0u;
}

// ---------- fp32 WMMA GEMM:  Out[M x P] = X[M x K] * W[P x K]^T + bias ----------
// Workgroup = 8 waves. Block owns one 64-column P-tile and 8 consecutive 16-row
// M-tiles (one per wave). The W tile is staged in LDS by async DMA
// (global_load_async_to_lds_b128), double-buffered in KCHUNK slabs, so B
// fragments are ds_load hits shared by all 8 waves; A fragments are
// software-pipelined global float2 loads.
// Frag layouts (V_WMMA_F32_16X16X4_F32):
//   A 16x4: lanes 0-15 -> M=lane, K pair {0,1}; lanes 16-31 -> K pair {2,3}.
//   B 4x16: mirrors A with N = lane&15.
//   C/D 16x16: VGPR r -> M = r + (lane>=16 ? 8 : 0), N = lane&15.
__global__ __launch_bounds__(256)
void gemm_bias_wmma(const float* __restrict__ X, const float* __restrict__ W,
                    const float* __restrict__ bias, float* __restrict__ Out,
                    int M, int K, int P) {
  __shared__ float wtile[2][64 * BSTRIDE];

  const int lane = threadIdx.x;
  const int wv   = threadIdx.y;                 // wave id 0..7
  const int tid  = wv * 32 + lane;
  const int ptiles = P >> 6;
  const int pt = blockIdx.x % ptiles;
  const int mg = blockIdx.x / ptiles;           // group of 8 m-tiles
  const int mt = mg * 8 + wv;
  const int mtiles = M >> 4;                    // M % 16 == 0 here
  const bool active = (mt < mtiles);
  const int mrow = (active ? mt : 0) * 16;      // clamp inactive waves to safe rows
  const int hi  = lane >> 4;
  const int l15 = lane & 15;
  const int cbase = pt * 64;

  const float* xrow = X + (size_t)(mrow + l15) * K + 2 * hi;

  // Stage one KCHUNK slab of the 64-column W tile into LDS buffer b.
  // 64 cols * 64 K floats = 1024 x 16B segments; 256 threads * 4 each.
  auto prefetch = [&](int kc, int b) {
#pragma unroll
    for (int s = tid; s < 1024; s += 256) {
      const int col = s >> 4;                   // 0..63
      const int q   = s & 15;                   // 16B segment within the row
      const float* g = W + (size_t)(cbase + col) * K + kc + q * 4;
      unsigned lds = (unsigned)(uintptr_t)&wtile[b][col * BSTRIDE + q * 4];
      async_copy16(g, lds);
    }
  };

  v8f c0 = {}, c1 = {}, c2 = {}, c3 = {};
  const int nchunks = K / KCHUNK;               // K % KCHUNK == 0 here (256/64)

  prefetch(0, 0);
  for (int c = 0; c < nchunks; ++c) {
    if (c + 1 < nchunks) {
      prefetch((c + 1) * KCHUNK, (c + 1) & 1);  // kick next slab first...
      wait_async_le();                          // ...then wait only for current slab
    } else {
      wait_async_0();
    }
    __syncthreads();                            // slab visible to all waves

    const int cb = c & 1;
    const float* achunk = xrow + c * KCHUNK;
    v2f a_nxt = *(const v2f*)(achunk);
#pragma unroll
    for (int kk = 0; kk < KCHUNK; kk += 4) {
      v2f a = a_nxt;
      if (kk + 4 < KCHUNK) a_nxt = *(const v2f*)(achunk + kk + 4);  // pipeline A
      const int kb = kk + 2 * hi;
      v2f b0 = *(const v2f*)&wtile[cb][(l15     ) * BSTRIDE + kb];
      v2f b1 = *(const v2f*)&wtile[cb][(l15 + 16) * BSTRIDE + kb];
      v2f b2 = *(const v2f*)&wtile[cb][(l15 + 32) * BSTRIDE + kb];
      v2f b3 = *(const v2f*)&wtile[cb][(l15 + 48) * BSTRIDE + kb];
      c0 = __builtin_amdgcn_wmma_f32_16x16x4_f32(false, a, false, b0, (short)0, c0, false, false);
      c1 = __builtin_amdgcn_wmma_f32_16x16x4_f32(false, a, false, b1, (short)0, c1, false, false);
      c2 = __builtin_amdgcn_wmma_f32_16x16x4_f32(false, a, false, b2, (short)0, c2, false, false);
      c3 = __builtin_amdgcn_wmma_f32_16x16x4_f32(false, a, false, b3, (short)0, c3, false, false);
    }
    __syncthreads();                            // done reading before next overwrite
  }

  if (active) {
    const int orow = mt * 16 + hi * 8;
    const float bb0 = bias[cbase + l15];
    const float bb1 = bias[cbase + l15 + 16];
    const float bb2 = bias[cbase + l15 + 32];
    const float bb3 = bias[cbase + l15 + 48];
#pragma unroll
    for (int r = 0; r < 8; ++r) {
      float* o = Out + (size_t)(orow + r) * P + cbase + l15;
      o[0]  = c0[r] + bb0;
      o[16] = c1[r] + bb1;
      o[32] = c2[r] + bb2;
      o[48] = c3[r] + bb3;
    }
  }
}

// ---------- GATv2 edge phase 1: logits e[eid][h] + segment max ----------
// One wave per edge. Self-loops appended logically for eid >= E.
__global__ void gat_logits(const float* __restrict__ xl, const float* __restrict__ xr,
                           const float* __restrict__ att, const int* __restrict__ ei,
                           float* __restrict__ elog, unsigned* __restrict__ mmax,
                           int E, int Etot, int dim, int hd) {
  const int eid = blockIdx.x * blockDim.y + threadIdx.y;
  if (eid >= Etot) return;
  const int lane = threadIdx.x;
  int srcn, dstn;
  if (eid < E) { srcn = ei[eid]; dstn = ei[E + eid]; }
  else         { srcn = dstn = eid - E; }
  const float* pl = xl + (size_t)srcn * dim;
  const float* pr = xr + (size_t)dstn * dim;
#pragma unroll
  for (int h = 0; h < NHEADS; ++h) {
    float p = 0.f;
    for (int j = lane; j < hd; j += 32) {
      float v = pl[h * hd + j] + pr[h * hd + j];
      v = v > 0.f ? v : SLOPE * v;               // leaky_relu
      p += v * att[h * hd + j];
    }
#pragma unroll
    for (int off = 16; off > 0; off >>= 1) p += __shfl_xor(p, off, 32);
    if (lane == 0) {
      elog[(size_t)eid * NHEADS + h] = p;
      atomicMax(&mmax[(size_t)dstn * NHEADS + h], flipf(p));
    }
  }
}

// ---------- GATv2 edge phase 2: w = exp(e - m[dst]); accumulate denom + numerator ----------
__global__ void gat_accum(const float* __restrict__ xl, const float* __restrict__ elog,
                          const unsigned* __restrict__ mmax, const int* __restrict__ ei,
                          float* __restrict__ den, float* __restrict__ acc,
                          int E, int Etot, int dim, int hd) {
  const int eid = blockIdx.x * blockDim.y + threadIdx.y;
  if (eid >= Etot) return;
  const int lane = threadIdx.x;
  int srcn, dstn;
  if (eid < E) { srcn = ei[eid]; dstn = ei[E + eid]; }
  else         { srcn = dstn = eid - E; }
  const float w0 = __expf(elog[(size_t)eid*NHEADS+0] - unflipf(mmax[(size_t)dstn*NHEADS+0]));
  const float w1 = __expf(elog[(size_t)eid*NHEADS+1] - unflipf(mmax[(size_t)dstn*NHEADS+1]));
  const float w2 = __expf(elog[(size_t)eid*NHEADS+2] - unflipf(mmax[(size_t)dstn*NHEADS+2]));
  const float w3 = __expf(elog[(size_t)eid*NHEADS+3] - unflipf(mmax[(size_t)dstn*NHEADS+3]));
  if (lane == 0) {
    atomicAdd(&den[(size_t)dstn*NHEADS+0], w0);
    atomicAdd(&den[(size_t)dstn*NHEADS+1], w1);
    atomicAdd(&den[(size_t)dstn*NHEADS+2], w2);
    atomicAdd(&den[(size_t)dstn*NHEADS+3], w3);
  }
  const float* pl = xl + (size_t)srcn * dim;
  float* pa = acc + (size_t)dstn * dim;
  for (int j = lane; j < hd; j += 32) atomicAdd(&pa[0*hd + j], w0 * pl[0*hd + j]);
  for (int j = lane; j < hd; j += 32) atomicAdd(&pa[1*hd + j], w1 * pl[1*hd + j]);
  for (int j = lane; j < hd; j += 32) atomicAdd(&pa[2*hd + j], w2 * pl[2*hd + j]);
  for (int j = lane; j < hd; j += 32) atomicAdd(&pa[3*hd + j], w3 * pl[3*hd + j]);
}

// ---------- finalize: acc = (acc / denom) + bias, optional ReLU ----------
__global__ void gat_finalize(float* __restrict__ acc, const float* __restrict__ den,
                             const float* __restrict__ bias, int N, int dim, int hd,
                             int do_relu) {
  long i = (long)blockIdx.x * blockDim.x + threadIdx.x;
  if (i >= (long)N * dim) return;
  int n = (int)(i / dim);
  int f = (int)(i - (long)n * dim);
  int h = f / hd;
  float v = acc[i] / den[(size_t)n * NHEADS + h] + bias[f];
  if (do_relu) v = fmaxf(v, 0.f);
  acc[i] = v;
}

// ---------- in-place LayerNorm over last dim (512), wave per node ----------
__global__ void layernorm_inplace(float* __restrict__ h, const float* __restrict__ gamma,
                                  const float* __restrict__ beta, int N) {
  const int n = blockIdx.x * blockDim.y + threadIdx.y;
  if (n >= N) return;
  const int lane = threadIdx.x;
  float* row = h + (size_t)n * OUTD;
  float s = 0.f, s2 = 0.f;
  for (int f = lane; f < OUTD; f += 32) { float v = row[f]; s += v; s2 += v * v; }
#pragma unroll
  for (int off = 16; off > 0; off >>= 1) { s += __shfl_xor(s, off, 32); s2 += __shfl_xor(s2, off, 32); }
  const float mean = s * (1.f / OUTD);
  const float var  = s2 * (1.f / OUTD) - mean * mean;
  const float rstd = rsqrtf(var + LN_EPS);
  for (int f = lane; f < OUTD; f += 32) {
    row[f] = gamma[f] * (row[f] - mean) * rstd + beta[f];
  }
}

// ---------- host ----------
static inline int cdiv(long a, long b) { return (int)((a + b - 1) / b); }

extern "C" void kernel_launch(void* const* d_in, const int* in_sizes, int n_in,
                              void* d_out, int out_size, void* d_ws, size_t ws_size,
                              hipStream_t stream) {
  const float* x     = (const float*)d_in[0];
  const int*   ei    = (const int*)  d_in[1];
  const float* Wl1   = (const float*)d_in[2];
  const float* bl1   = (const float*)d_in[3];
  const float* Wr1   = (const float*)d_in[4];
  const float* br1   = (const float*)d_in[5];
  const float* att1  = (const float*)d_in[6];
  const float* bias1 = (const float*)d_in[7];
  const float* Wl2   = (const float*)d_in[8];
  const float* bl2   = (const float*)d_in[9];
  const float* Wr2   = (const float*)d_in[10];
  const float* br2   = (const float*)d_in[11];
  const float* att2  = (const float*)d_in[12];
  const float* bias2 = (const float*)d_in[13];
  const float* gamma = (const float*)d_in[14];
  const float* beta  = (const float*)d_in[15];

  const int N    = in_sizes[0] / F_IN;   // 10000
  const int E    = in_sizes[1] / 2;      // 320000
  const int Etot = E + N;                // self-loops appended

  // workspace layout (floats)
  float* ws = (float*)d_ws;
  size_t o = 0;
  float*    XL   = ws + o; o += (size_t)N * OUTD;   // src transform (max width 512)
  float*    XR   = ws + o; o += (size_t)N * OUTD;   // dst transform
  float*    H1   = ws + o; o += (size_t)N * HID;    // layer-1 aggregated output
  float*    ELOG = ws + o; o += (size_t)Etot * NHEADS;
  unsigned* MMAX = (unsigned*)(ws + o); o += (size_t)N * NHEADS;
  float*    DEN  = ws + o; o += (size_t)N * NHEADS;
  float*    H2   = (float*)d_out;                   // layer-2 accumulator == output

  const dim3 wblk(32, 8);                 // 8 waves / workgroup (wave32)
  const int  ethr = 256;
  const int  mtiles  = N / 16;            // 625
  const int  mgroups = cdiv(mtiles, 8);   // 8 m-tiles (one per wave) per block

  // ===== layer 1 =====
  fill_zero_u32<<<cdiv((long)N * HID, ethr), ethr, 0, stream>>>((unsigned*)H1, (long)N * HID);
  fill_zero_u32<<<cdiv((long)N * NHEADS, ethr), ethr, 0, stream>>>(MMAX, (long)N * NHEADS);
  fill_zero_u32<<<cdiv((long)N * NHEADS, ethr), ethr, 0, stream>>>((unsigned*)DEN, (long)N * NHEADS);

  gemm_bias_wmma<<<mgroups * (HID / 64), wblk, 0, stream>>>(x, Wl1, bl1, XL, N, F_IN, HID);
  gemm_bias_wmma<<<mgroups * (HID / 64), wblk, 0, stream>>>(x, Wr1, br1, XR, N, F_IN, HID);

  gat_logits<<<cdiv(Etot, 8), wblk, 0, stream>>>(XL, XR, att1, ei, ELOG, MMAX, E, Etot, HID, HID / NHEADS);
  gat_accum <<<cdiv(Etot, 8), wblk, 0, stream>>>(XL, ELOG, MMAX, ei, DEN, H1, E, Etot, HID, HID / NHEADS);
  gat_finalize<<<cdiv((long)N * HID, ethr), ethr, 0, stream>>>(H1, DEN, bias1, N, HID, HID / NHEADS, 1);

  // ===== layer 2 =====
  fill_zero_u32<<<cdiv((long)N * OUTD, ethr), ethr, 0, stream>>>((unsigned*)H2, (long)N * OUTD);
  fill_zero_u32<<<cdiv((long)N * NHEADS, ethr), ethr, 0, stream>>>(MMAX, (long)N * NHEADS);
  fill_zero_u32<<<cdiv((long)N * NHEADS, ethr), ethr, 0, stream>>>((unsigned*)DEN, (long)N * NHEADS);

  gemm_bias_wmma<<<mgroups * (OUTD / 64), wblk, 0, stream>>>(H1, Wl2, bl2, XL, N, HID, OUTD);
  gemm_bias_wmma<<<mgroups * (OUTD / 64), wblk, 0, stream>>>(H1, Wr2, br2, XR, N, HID, OUTD);

  gat_logits<<<cdiv(Etot, 8), wblk, 0, stream>>>(XL, XR, att2, ei, ELOG, MMAX, E, Etot, OUTD, OUTD / NHEADS);
  gat_accum <<<cdiv(Etot, 8), wblk, 0, stream>>>(XL, ELOG, MMAX, ei, DEN, H2, E, Etot, OUTD, OUTD / NHEADS);
  gat_finalize<<<cdiv((long)N * OUTD, ethr), ethr, 0, stream>>>(H2, DEN, bias2, N, OUTD, OUTD / NHEADS, 0);

  // ===== LayerNorm (in place on d_out) =====
  layernorm_inplace<<<cdiv(N, 8), wblk, 0, stream>>>(H2, gamma, beta, N);
}